// CLUB_27479200759809
// MI455X (gfx1250) — compile-verified
//
#include <hip/hip_runtime.h>
#include <math.h>

#define NROWS 512
#define DCOLS 256
#define EPS_F 1e-7f

typedef __attribute__((ext_vector_type(2))) float v2f;
typedef __attribute__((ext_vector_type(8))) float v8f;

// ---------------------------------------------------------------------------
// Kernel 1: column moments of h:  S1[d] = sum_j h[j,d],  S2[d] = sum_j h[j,d]^2
// One block of 256 threads; thread d owns column d -> fully coalesced rows.
// ---------------------------------------------------------------------------
__global__ void club_colstats(const float* __restrict__ h, float* __restrict__ ws) {
    const int d = threadIdx.x;              // 0..255
    float s1 = 0.0f, s2 = 0.0f;
    for (int j = 0; j < NROWS; ++j) {
        const float v = h[j * DCOLS + d];
        s1 += v;
        s2 = fmaf(v, v, s2);
    }
    ws[d]         = s1;
    ws[DCOLS + d] = s2;
}

// ---------------------------------------------------------------------------
// Kernel 2: fused term computation + WMMA row reduction.
//   term[i,d] = -0.5 * inv * ((mu-h)^2 - (S2/N - 2*mu*S1/N + mu^2))
//   rowsum[i] = term[i,:] @ ones  via V_WMMA_F32_16X16X4_F32 (B = ones).
// One wave per 16-row tile; 32 tiles total. A-matrix f32 16x4 lane layout:
// lanes 0-15 -> K={0,1}, lanes 16-31 -> K={2,3} for rows M=0..15.
// ---------------------------------------------------------------------------
__global__ void club_wmma_reduce(const float* __restrict__ mu,
                                 const float* __restrict__ logvar,
                                 const float* __restrict__ h,
                                 const float* __restrict__ ws,
                                 float* __restrict__ partials) {
    const int lane = threadIdx.x & 31;
    const int wave = threadIdx.x >> 5;
    const int tile = blockIdx.x * 4 + wave;   // 16-row tile index, 0..31
    const int half = lane >> 4;               // 0 -> K pair {0,1}, 1 -> {2,3}
    const int m    = lane & 15;               // row within tile
    const int row  = tile * 16 + m;

    const float invN = 1.0f / (float)NROWS;

    const float* __restrict__ murow = mu     + row * DCOLS;
    const float* __restrict__ lvrow = logvar + row * DCOLS;
    const float* __restrict__ hrow  = h      + row * DCOLS;
    const float* __restrict__ s1col = ws;
    const float* __restrict__ s2col = ws + DCOLS;

    v8f c = {};
    v2f b;  b[0] = 1.0f;  b[1] = 1.0f;        // ones matrix: layout-invariant

    for (int kc = 0; kc < DCOLS / 4; ++kc) {
        const int kb = kc * 4 + half * 2;     // even -> 8B aligned float2 loads
        const float2 m2 = *(const float2*)(murow + kb);
        const float2 l2 = *(const float2*)(lvrow + kb);
        const float2 h2 = *(const float2*)(hrow  + kb);

        v2f a;
#pragma unroll
        for (int e = 0; e < 2; ++e) {
            const float mv = e ? m2.y : m2.x;
            const float lv = e ? l2.y : l2.x;
            const float hv = e ? h2.y : h2.x;
            const float inv  = 1.0f / (expf(lv) + EPS_F);
            const float diff = mv - hv;
            const float s1 = s1col[kb + e];
            const float s2 = s2col[kb + e];
            // neg_mean = S2/N - 2*mu*S1/N + mu^2
            const float negmean = fmaf(-2.0f * mv, s1 * invN, s2 * invN) + mv * mv;
            a[e] = -0.5f * inv * (diff * diff - negmean);
        }
        // D = A(16x4) * ones(4x16) + C : accumulates row sums in every column.
        c = __builtin_amdgcn_wmma_f32_16x16x4_f32(
                /*neg_a=*/false, a, /*neg_b=*/false, b,
                /*c_mod=*/(short)0, c, /*reuse_a=*/false, /*reuse_b=*/false);
    }

    // Each lane's 8 C regs hold rows {m_base..m_base+7} of one column.
    float s = c[0] + c[1] + c[2] + c[3] + c[4] + c[5] + c[6] + c[7];
    // Fold rows 0-7 (lanes 0-15) with rows 8-15 (lanes 16-31).
    s += __shfl_xor(s, 16, 32);
    if (lane == 0) partials[tile] = s;
}

// ---------------------------------------------------------------------------
// Kernel 3: fold 32 tile partials, scale by 1/N.
// ---------------------------------------------------------------------------
__global__ void club_finalize(const float* __restrict__ partials,
                              float* __restrict__ out) {
    float v = partials[threadIdx.x];          // 32 threads
    for (int off = 16; off > 0; off >>= 1) v += __shfl_xor(v, off, 32);
    if (threadIdx.x == 0) out[0] = v * (1.0f / (float)NROWS);
}

extern "C" void kernel_launch(void* const* d_in, const int* in_sizes, int n_in,
                              void* d_out, int out_size, void* d_ws, size_t ws_size,
                              hipStream_t stream) {
    const float* mu     = (const float*)d_in[0];
    const float* logvar = (const float*)d_in[1];
    const float* h      = (const float*)d_in[2];
    float* out = (float*)d_out;
    float* ws  = (float*)d_ws;                // [0,256): S1, [256,512): S2
    float* partials = ws + 2 * DCOLS;         // [512,544): tile partials

    club_colstats<<<1, DCOLS, 0, stream>>>(h, ws);
    club_wmma_reduce<<<8, 128, 0, stream>>>(mu, logvar, h, ws, partials);
    club_finalize<<<1, 32, 0, stream>>>(partials, out);
}